// ModelLossSemsegGatedCRF3D_88235808129166
// MI455X (gfx1250) — compile-verified
//
#include <hip/hip_runtime.h>

typedef __attribute__((ext_vector_type(2))) float v2f;
typedef __attribute__((ext_vector_type(4))) float v4f;
typedef __attribute__((ext_vector_type(8))) float v8f;

#define HH 64
#define WW 64
#define DD 32
#define HWD (HH * WW * DD)     // 131072
#define SIGMA_XY 5.0f
#define SIGMA_IMG 0.1f
// sqrt(log2(e)): features scaled by this so the WMMA dot product and the
// stored norms are already in exp2-domain (saves a v_mul per exp).
#define SQRT_LOG2E 1.2011224087864498f

#if __has_builtin(__builtin_amdgcn_exp2f)
#define EXP2F(x) __builtin_amdgcn_exp2f(x)
#else
#define EXP2F(x) exp2f(x)
#endif

// ---------------------------------------------------------------------------
// Kernel 1: build 4-dim features (mesh/sigma_xy, sample/sigma_img), scaled by
// sqrt(log2 e), in AoS layout feats[v][4]; plus a packed 16-byte side-record
// aux[v] = {-0.5*||f'||^2, y0, y1, 0} (norm pre-negated/halved, exp2-domain)
// so the main kernel fetches all per-voxel scalars with one b128 load.
// Also zeroes the scalar output (stream-ordered before the atomics).
// ---------------------------------------------------------------------------
__global__ __launch_bounds__(256) void gatedcrf_feats_kernel(
    const float* __restrict__ y,        // (1,2,H,W,D)
    const float* __restrict__ sample,   // (1,1,H,W,D)
    const float* __restrict__ spacing,  // (3,1)
    float* __restrict__ feats,          // [HWD][4]
    v4f* __restrict__ aux,              // [HWD] {-0.5*nrm', y0, y1, 0}
    float* __restrict__ out) {
  int v = blockIdx.x * blockDim.x + threadIdx.x;
  if (v == 0) out[0] = 0.0f;
  if (v >= HWD) return;
  int d = v & (DD - 1);
  int w = (v / DD) & (WW - 1);
  int h = v / (DD * WW);
  float f0 = spacing[0] * (float)h * (SQRT_LOG2E / SIGMA_XY);
  float f1 = spacing[1] * (float)w * (SQRT_LOG2E / SIGMA_XY);
  float f2 = spacing[2] * (float)d * (SQRT_LOG2E / SIGMA_XY);
  float f3 = sample[v] * (SQRT_LOG2E / SIGMA_IMG);
  feats[4 * v + 0] = f0;
  feats[4 * v + 1] = f1;
  feats[4 * v + 2] = f2;
  feats[4 * v + 3] = f3;
  v4f a;
  a.x = -0.5f * (f0 * f0 + f1 * f1 + f2 * f2 + f3 * f3);
  a.y = y[v];
  a.z = y[HWD + v];
  a.w = 0.0f;
  aux[v] = a;
}

// ---------------------------------------------------------------------------
// Kernel 2: one wave per tile of 16 centers (fixed h,d; w in [w0,w0+16)).
// For each of the 21 (dy,dz) offsets and two 16-wide neighbor column tiles,
// one V_WMMA_F32_16X16X4_F32 produces all 16x16 pairwise 4-dim feature dots;
// k = exp2(dot' + n'_p + n'_q) on in-window taps, accumulate k*(1 - <y_p,y_q>).
// Fully branchless: OOB neighbors are redirected to a valid address and their
// values zeroed (zero-pad semantics); tap masks are v_cndmask selects.
// ---------------------------------------------------------------------------
__global__ __launch_bounds__(256) void gatedcrf_main_kernel(
    const float* __restrict__ feats,   // [HWD][4]
    const v4f* __restrict__ aux,       // [HWD]
    float* __restrict__ out) {
  const int lane = threadIdx.x & 31;
  const int wave = threadIdx.x >> 5;
  const int tile = blockIdx.x * 8 + wave;  // 8192 tiles total
  const int col  = lane & 15;              // B column / A row index
  const int hi   = lane >> 4;              // K half (K = 2*hi, 2*hi+1)

  const int wt = tile & 3;
  const int d  = (tile >> 2) & (DD - 1);
  const int h  = tile >> 7;
  const int w0 = wt * 16;

  // A matrix: 16 center feature rows (M = col), K-half per lane.
  const int cvA = (h * WW + (w0 + col)) * DD + d;
  const v2f A = *(const v2f*)(feats + 4 * cvA + 2 * hi);

  // Per-row center scalars for this lane's 8 D-rows (m = 8*hi + r).
  float nph[8], y0p[8], y1p[8];
#pragma unroll
  for (int r = 0; r < 8; ++r) {
    int m  = 8 * hi + r;
    int cv = (h * WW + (w0 + m)) * DD + d;
    v4f ac = aux[cv];
    nph[r] = ac.x;        // already -0.5*||f'||^2
    y0p[r] = ac.y;
    y1p[r] = ac.z;
  }

  // Window masks depend only on (t, r, col): dx = col - 3 + 16*t - m.
  // Precompute with / without the center tap (dx==0) excluded.
  float mwin[2][8], mwin_nc[2][8];
#pragma unroll
  for (int t = 0; t < 2; ++t) {
#pragma unroll
    for (int r = 0; r < 8; ++r) {
      int m  = 8 * hi + r;
      int dx = col - 3 + 16 * t - m;
      bool win = (unsigned)(dx + 3) <= 6u;
      mwin[t][r]    = win ? 1.0f : 0.0f;
      mwin_nc[t][r] = (win && dx != 0) ? 1.0f : 0.0f;
    }
  }

  const v8f zeroC = {};
  float acc = 0.0f;

#pragma unroll
  for (int dy = -3; dy <= 3; ++dy) {
    const int qh = h + dy;
#pragma unroll
    for (int dz = -1; dz <= 1; ++dz) {
      const int qd = d + dz;
      const bool rowOK = ((unsigned)qh < (unsigned)HH) &&
                         ((unsigned)qd < (unsigned)DD);
      const bool centerIter = (dy == 0) && (dz == 0);
#pragma unroll
      for (int t = 0; t < 2; ++t) {
        const int qw = w0 - 3 + 16 * t + col;  // this lane's B column position
        const bool ok = rowOK && ((unsigned)qw < (unsigned)WW);
        const float okf = ok ? 1.0f : 0.0f;
        const int qv = ok ? (qh * WW + qw) * DD + qd : cvA;  // clamped addr

        v2f Bf = *(const v2f*)(feats + 4 * qv + 2 * hi);
        v4f aq = aux[qv];
        Bf.x *= okf;                 // zero-pad semantics: f_q = 0 when OOB
        Bf.y *= okf;
        const float nqh = aq.x * okf;  // already -0.5*||f'||^2
        const float yq0 = aq.y * okf;
        const float yq1 = aq.z * okf;

        // Pre-add per-row + per-column norm terms once per tile.
        float nn[8];
#pragma unroll
        for (int r = 0; r < 8; ++r) nn[r] = nph[r] + nqh;

        // D[m][n] = f'_center(m) . f'_neighbor(n)  (16x16x4 f32 WMMA)
        v8f Dv = __builtin_amdgcn_wmma_f32_16x16x4_f32(
            /*neg_a=*/false, A, /*neg_b=*/false, Bf,
            /*c_mod=*/(short)0, zeroC, /*reuse_a=*/false, /*reuse_b=*/false);

#pragma unroll
        for (int r = 0; r < 8; ++r) {
          const float tapf = centerIter ? mwin_nc[t][r] : mwin[t][r];
          const float k    = EXP2F(Dv[r] + nn[r]);        // = exp(-0.5*dist^2)
          const float oms  = 1.0f - (y0p[r] * yq0 + y1p[r] * yq1);
          acc = fmaf(tapf * k, oms, acc);
        }
      }
    }
  }

  // wave32 reduction, then one atomic per wave (loss normalized by L).
  for (int off = 16; off > 0; off >>= 1) acc += __shfl_down(acc, off, 32);
  if (lane == 0) atomicAdd(out, acc * (1.0f / (float)HWD));
}

// ---------------------------------------------------------------------------
extern "C" void kernel_launch(void* const* d_in, const int* in_sizes, int n_in,
                              void* d_out, int out_size, void* d_ws, size_t ws_size,
                              hipStream_t stream) {
  const float* y       = (const float*)d_in[0];  // y_hat_softmax (1,2,64,64,32)
  const float* sample  = (const float*)d_in[1];  // (1,1,64,64,32)
  const float* spacing = (const float*)d_in[2];  // (3,1)
  float* feats = (float*)d_ws;                   // 4*HWD floats (2 MB)
  v4f*   aux   = (v4f*)((char*)d_ws + 4 * HWD * sizeof(float));  // 2 MB
  float* out   = (float*)d_out;                  // scalar loss

  gatedcrf_feats_kernel<<<HWD / 256, 256, 0, stream>>>(y, sample, spacing,
                                                       feats, aux, out);
  // 8192 tiles (H*D*W/16), 8 waves per 256-thread block -> 1024 blocks
  gatedcrf_main_kernel<<<1024, 256, 0, stream>>>(feats, aux, out);
}